// TensorProductUniform3x1d_52879637348702
// MI455X (gfx1250) — compile-verified
//
#include <hip/hip_runtime.h>
#include <stdint.h>

// Problem constants (from the reference)
#define NSEG   4
#define U      128
#define ROW_F  (NSEG * U)        // 512 floats per batch row
#define ROW_B  (ROW_F * 4)       // 2048 bytes per batch row
#define CH_ROWS 8                // batch rows staged per chunk
#define CH_F   (CH_ROWS * ROW_F) // 4096 floats per array per chunk
#define CH_B   (CH_F * 4)        // 16384 bytes per array per chunk
#define BLK    256               // 8 wave32 per block

typedef __attribute__((ext_vector_type(4))) float v4f;

// ---------------------------------------------------------------------------
// Prep kernel: collapse the (≤64-entry) sparse path list into a dense
// W[j][a][c] bilinear weight tensor in d_ws. Runs once, trivially cheap.
// ---------------------------------------------------------------------------
__global__ void build_w_kernel(const float* __restrict__ coef,
                               const int* __restrict__ pidx,
                               float* __restrict__ W, int npath) {
  int e = threadIdx.x;
  if (e < 64) {
    int j = e >> 4, a = (e >> 2) & 3, c = e & 3;
    float s = 0.f;
    for (int p = 0; p < npath; ++p) {
      if (pidx[3 * p + 0] == a && pidx[3 * p + 1] == c && pidx[3 * p + 2] == j)
        s += coef[p];
    }
    W[e] = s;
  }
}

// ---------------------------------------------------------------------------
// CDNA5 async global->LDS 16B copy (per-lane), GVS addressing:
//   copies 16 bytes from global address base plus offset voff to LDS byte address lds_byte (tracked by ASYNCcnt)
// ---------------------------------------------------------------------------
__device__ __forceinline__ void async_b128(uint32_t lds_byte,
                                           const float* base, uint32_t voff) {
  asm volatile("global_load_async_to_lds_b128 %0, %1, %2 offset:0"
               :: "v"(lds_byte), "v"(voff), "s"(base)
               : "memory");
}

#define WAIT_ASYNC(n) asm volatile("s_wait_asynccnt " #n ::: "memory")

// ---------------------------------------------------------------------------
// Main streaming kernel: double-buffered async DMA pipeline.
// ---------------------------------------------------------------------------
__global__ void __launch_bounds__(BLK)
tp_uniform_kernel(const float* __restrict__ x0, const float* __restrict__ x1,
                  const float* __restrict__ W, float* __restrict__ out, int B) {
  __shared__ float smem[2][2][CH_F];   // [buffer][array(x0/x1)][chunk floats]

  const int tid = threadIdx.x;
  const int r   = tid >> 5;            // local row within chunk   (0..7)
  const int cc  = tid & 31;            // float4 column within row (0..31)

  const uint32_t nchunks = (uint32_t)((B + CH_ROWS - 1) / CH_ROWS);
  const uint32_t limit   = (uint32_t)B * (uint32_t)ROW_B - 16u; // last legal 16B load

  // LDS byte offset of smem base: low 32 bits of the generic pointer are the
  // LDS byte address.
  const uint32_t lds0 = (uint32_t)(uintptr_t)(&smem[0][0][0]);

  // Uniform weights: kernarg pointer + literal offsets -> scalar loads.
  float w[64];
#pragma unroll
  for (int k = 0; k < 64; ++k) w[k] = W[k];

  // Issue one chunk (8 rows of x0 + x1) into LDS buffer `buf`:
  // 8 async b128 per thread -> ASYNCcnt += 8 per wave.
  auto issue = [&](uint32_t chunk, uint32_t buf) {
    const uint32_t gbase = chunk * (uint32_t)CH_B;
#pragma unroll
    for (int k = 0; k < 4; ++k) {
      uint32_t fi  = (uint32_t)tid + (uint32_t)k * BLK;   // b128 slot in chunk
      uint32_t off = gbase + fi * 16u;
      if (off > limit) off = limit;                       // clamp tail (safe)
      uint32_t l0 = lds0 + (buf * 2u + 0u) * (uint32_t)CH_B + fi * 16u;
      uint32_t l1 = lds0 + (buf * 2u + 1u) * (uint32_t)CH_B + fi * 16u;
      async_b128(l0, x0, off);
      async_b128(l1, x1, off);
    }
  };

  uint32_t c   = blockIdx.x;
  uint32_t buf = 0;
  if (c < nchunks) issue(c, 0);        // prologue prefetch

  for (; c < nchunks; c += gridDim.x) {
    const uint32_t cn = c + gridDim.x;
    if (cn < nchunks) {
      issue(cn, buf ^ 1u);             // prefetch next chunk into other buffer
      WAIT_ASYNC(8);                   // retire current chunk; next stays in flight
    } else {
      WAIT_ASYNC(0);
    }
    __syncthreads();                   // LDS data visible to all waves

    // ---- compute: one (row, float4-column) per thread ----
    const float* s0 = &smem[buf][0][r * ROW_F];
    const float* s1 = &smem[buf][1][r * ROW_F];
    v4f a0[NSEG], a1[NSEG];
#pragma unroll
    for (int a = 0; a < NSEG; ++a) {
      a0[a] = *(const v4f*)&s0[a * U + cc * 4];   // ds_load_b128
      a1[a] = *(const v4f*)&s1[a * U + cc * 4];
    }
    v4f acc[NSEG];
    const v4f vzero = {0.f, 0.f, 0.f, 0.f};
#pragma unroll
    for (int j = 0; j < NSEG; ++j) acc[j] = vzero;
#pragma unroll
    for (int a = 0; a < NSEG; ++a) {
#pragma unroll
      for (int b = 0; b < NSEG; ++b) {
        v4f p = a0[a] * a1[b];
#pragma unroll
        for (int j = 0; j < NSEG; ++j)
          acc[j] += w[j * 16 + a * 4 + b] * p;    // SGPR-weight FMAs
      }
    }

    const uint32_t row = c * (uint32_t)CH_ROWS + (uint32_t)r;
    if (row < (uint32_t)B) {
#pragma unroll
      for (int j = 0; j < NSEG; ++j)
        *(v4f*)&out[row * ROW_F + j * U + cc * 4] = acc[j];  // global_store_b128
    }

    __syncthreads();                   // all reads of `buf` done before reuse
    buf ^= 1u;
  }
}

// ---------------------------------------------------------------------------
extern "C" void kernel_launch(void* const* d_in, const int* in_sizes, int n_in,
                              void* d_out, int out_size, void* d_ws, size_t ws_size,
                              hipStream_t stream) {
  const float* x0   = (const float*)d_in[0];
  const float* x1   = (const float*)d_in[1];
  const float* coef = (const float*)d_in[2];
  const int*   pidx = (const int*)d_in[3];
  float*       W    = (float*)d_ws;
  float*       out  = (float*)d_out;

  const int B     = in_sizes[0] / ROW_F;
  const int npath = in_sizes[2];

  build_w_kernel<<<1, 64, 0, stream>>>(coef, pidx, W, npath);

  const uint32_t nchunks = (uint32_t)((B + CH_ROWS - 1) / CH_ROWS);
  if (nchunks == 0) return;
  const uint32_t blocks = nchunks < 2048u ? nchunks : 2048u;
  tp_uniform_kernel<<<blocks, BLK, 0, stream>>>(x0, x1, W, out, B);
}